// ChamferDistanceL2_withnormal_normalindex_55482387530095
// MI455X (gfx1250) — compile-verified
//
#include <hip/hip_runtime.h>
#include <hip/hip_bf16.h>

typedef __attribute__((ext_vector_type(2))) float v2f;
typedef __attribute__((ext_vector_type(8))) float v8f;

#define BB 4
#define NN 8192
#define MM 8192
#define EPSN 1e-12f
#define FLT_BIG 3.402823466e38f

__device__ __forceinline__ float3 norm3(float x, float y, float z) {
    float n = sqrtf(x * x + y * y + z * z);
    float inv = 1.0f / fmaxf(n, EPSN);
    return make_float3(x * inv, y * inv, z * inv);
}

// Build augmented 8-D query/database forms so that Q . D == squared 6-D distance.
// Q = {-2x,-2y,-2z,-2nx,-2ny,-2nz, sq, 1}   D = {x,y,z,nx,ny,nz, 1, sq}
__global__ void prep_kernel(const float* __restrict__ xyz,
                            const float* __restrict__ nrm,
                            float* __restrict__ Q, float* __restrict__ D,
                            int total) {
    int i = blockIdx.x * blockDim.x + threadIdx.x;
    if (i >= total) return;
    float x = xyz[i * 3 + 0], y = xyz[i * 3 + 1], z = xyz[i * 3 + 2];
    float3 nv = norm3(nrm[i * 3 + 0], nrm[i * 3 + 1], nrm[i * 3 + 2]);
    float sq = x * x + y * y + z * z + nv.x * nv.x + nv.y * nv.y + nv.z * nv.z;
    float* q = Q + (size_t)i * 8;
    float* d = D + (size_t)i * 8;
    q[0] = -2.f * x;    q[1] = -2.f * y;    q[2] = -2.f * z;
    q[3] = -2.f * nv.x; q[4] = -2.f * nv.y; q[5] = -2.f * nv.z;
    q[6] = sq;          q[7] = 1.f;
    d[0] = x;    d[1] = y;    d[2] = z;
    d[3] = nv.x; d[4] = nv.y; d[5] = nv.z;
    d[6] = 1.f;  d[7] = sq;
}

// One wave owns 16 query rows, scans the whole database via f32 WMMA tiles.
// A-matrix 16x4 layout: lanes 0-15 hold K=0,1 in v[0],v[1]; lanes 16-31 hold K=2,3.
// C layout: VGPR r, lanes 0-15 -> M=r, N=lane; lanes 16-31 -> M=8+r, N=lane-16.
template <int WAVES>
__global__ void nn_kernel(const float* __restrict__ Q,
                          const float* __restrict__ D,
                          int* __restrict__ idxOut, int nQ, int nD) {
    const int wave = threadIdx.x >> 5;
    const int lane = threadIdx.x & 31;
    const int half = lane >> 4;
    const int l16  = lane & 15;
    const int b = blockIdx.y;
    const int rowBase = (blockIdx.x * WAVES + wave) * 16;
    const int k0 = half * 2;

    const float* qb = Q + (size_t)b * nQ * 8;
    const float* db = D + (size_t)b * nD * 8;

    const int qrow = rowBase + l16;
    v2f a_lo = *(const v2f*)(qb + (size_t)qrow * 8 + k0);       // K = k0, k0+1
    v2f a_hi = *(const v2f*)(qb + (size_t)qrow * 8 + k0 + 4);   // K = k0+4, k0+5

    float best[8];
    int bidx[8];
#pragma unroll
    for (int r = 0; r < 8; ++r) { best[r] = FLT_BIG; bidx[r] = 0; }

    for (int ct = 0; ct < nD; ct += 16) {
        const int col = ct + l16;
        v2f b_lo = *(const v2f*)(db + (size_t)col * 8 + k0);
        v2f b_hi = *(const v2f*)(db + (size_t)col * 8 + k0 + 4);
        v8f c = {0.f, 0.f, 0.f, 0.f, 0.f, 0.f, 0.f, 0.f};
        c = __builtin_amdgcn_wmma_f32_16x16x4_f32(false, a_lo, false, b_lo,
                                                  (short)0, c, false, false);
        c = __builtin_amdgcn_wmma_f32_16x16x4_f32(false, a_hi, false, b_hi,
                                                  (short)0, c, false, false);
#pragma unroll
        for (int r = 0; r < 8; ++r) {
            float v = c[r];
            if (v < best[r]) { best[r] = v; bidx[r] = col; }
        }
    }

    // reduce over the 16 columns held across each 16-lane half (xor stays in-half)
#pragma unroll
    for (int off = 1; off < 16; off <<= 1) {
#pragma unroll
        for (int r = 0; r < 8; ++r) {
            float ov = __shfl_xor(best[r], off, 32);
            int   oi = __shfl_xor(bidx[r], off, 32);
            if (ov < best[r] || (ov == best[r] && oi < bidx[r])) {
                best[r] = ov; bidx[r] = oi;
            }
        }
    }

    if (l16 == 0) {
#pragma unroll
        for (int r = 0; r < 8; ++r)
            idxOut[(size_t)b * nQ + rowBase + half * 8 + r] = bidx[r];
    }
}

__global__ void zero_out_kernel(float* __restrict__ out) {
    out[0] = 0.f;
    out[1] = 0.f;
}

// Gather by NN index, accumulate xyz-chamfer and sign-invariant normal terms.
__global__ void loss_kernel(const float* __restrict__ xyzQ,
                            const float* __restrict__ nrmQ,
                            const float* __restrict__ xyzD,
                            const float* __restrict__ nrmD,
                            const int* __restrict__ idx, int nQ, int nD,
                            float scale, float* __restrict__ out) {
    __shared__ float sx[256];
    __shared__ float sn[256];
    const int i = blockIdx.x * blockDim.x + threadIdx.x;  // over BB*nQ (exact)
    const int b = i / nQ;
    const int q = i - b * nQ;
    const int j = idx[i];

    const float* xq = xyzQ + ((size_t)b * nQ + q) * 3;
    const float* xd = xyzD + ((size_t)b * nD + j) * 3;
    float dx = xq[0] - xd[0], dy = xq[1] - xd[1], dz = xq[2] - xd[2];
    float xs = dx * dx + dy * dy + dz * dz;

    const float* aq = nrmQ + ((size_t)b * nQ + q) * 3;
    const float* ad = nrmD + ((size_t)b * nD + j) * 3;
    float3 a = norm3(aq[0], aq[1], aq[2]);
    float3 t = norm3(ad[0], ad[1], ad[2]);
    float m1 = (a.x - t.x) * (a.x - t.x) + (a.y - t.y) * (a.y - t.y) +
               (a.z - t.z) * (a.z - t.z);
    float p1 = (a.x + t.x) * (a.x + t.x) + (a.y + t.y) * (a.y + t.y) +
               (a.z + t.z) * (a.z + t.z);
    float ns = fminf(m1, p1);

    sx[threadIdx.x] = xs;
    sn[threadIdx.x] = ns;
    __syncthreads();
    for (int s = 128; s > 0; s >>= 1) {
        if (threadIdx.x < s) {
            sx[threadIdx.x] += sx[threadIdx.x + s];
            sn[threadIdx.x] += sn[threadIdx.x + s];
        }
        __syncthreads();
    }
    if (threadIdx.x == 0) {
        atomicAdd(&out[0], sx[0] * scale);
        atomicAdd(&out[1], sn[0] * scale);
    }
}

extern "C" void kernel_launch(void* const* d_in, const int* in_sizes, int n_in,
                              void* d_out, int out_size, void* d_ws, size_t ws_size,
                              hipStream_t stream) {
    (void)in_sizes; (void)n_in; (void)out_size; (void)ws_size;

    const float* xyz1 = (const float*)d_in[0];  // [B,N,3]
    const float* xyz2 = (const float*)d_in[1];  // [B,M,3]
    const float* nrm1 = (const float*)d_in[2];  // normal_rebuild [B,N,3]
    const float* nrm2 = (const float*)d_in[3];  // normal_gt      [B,M,3]
    float* out = (float*)d_out;

    // workspace layout
    char* ws = (char*)d_ws;
    const size_t augFloats = (size_t)BB * NN * 8;  // N == M
    float* Q1 = (float*)ws;                ws += augFloats * sizeof(float);
    float* D1 = (float*)ws;                ws += augFloats * sizeof(float);
    float* Q2 = (float*)ws;                ws += augFloats * sizeof(float);
    float* D2 = (float*)ws;                ws += augFloats * sizeof(float);
    int* idx1 = (int*)ws;                  ws += (size_t)BB * NN * sizeof(int);
    int* idx2 = (int*)ws;

    zero_out_kernel<<<1, 1, 0, stream>>>(out);

    const int tot1 = BB * NN, tot2 = BB * MM;
    prep_kernel<<<(tot1 + 255) / 256, 256, 0, stream>>>(xyz1, nrm1, Q1, D1, tot1);
    prep_kernel<<<(tot2 + 255) / 256, 256, 0, stream>>>(xyz2, nrm2, Q2, D2, tot2);

    constexpr int WAVES = 4;
    dim3 gridA(NN / 16 / WAVES, BB);
    dim3 gridB(MM / 16 / WAVES, BB);
    nn_kernel<WAVES><<<gridA, WAVES * 32, 0, stream>>>(Q1, D2, idx1, NN, MM);
    nn_kernel<WAVES><<<gridB, WAVES * 32, 0, stream>>>(Q2, D1, idx2, MM, NN);

    loss_kernel<<<tot1 / 256, 256, 0, stream>>>(xyz1, nrm1, xyz2, nrm2, idx1,
                                                NN, MM, 1.0f / tot1, out);
    loss_kernel<<<tot2 / 256, 256, 0, stream>>>(xyz2, nrm2, xyz1, nrm1, idx2,
                                                MM, NN, 1.0f / tot2, out);
}